// GatedMultiScaleStripAdapterV4_17557826306546
// MI455X (gfx1250) — compile-verified
//
#include <hip/hip_runtime.h>
#include <hip/hip_bf16.h>

typedef __attribute__((ext_vector_type(2))) float v2f;
typedef __attribute__((ext_vector_type(8))) float v8f;

#define WMMA4(A, Bv, Cc) \
  __builtin_amdgcn_wmma_f32_16x16x4_f32(false, (A), false, (Bv), (short)0, (Cc), false, false)

#define CC   46
#define HH   64
#define WWi  64
#define HWp  4096
#define BB   8
#define KT   23    // taps per strip
#define C6   276   // 6*C
#define K9   414   // C*9
#define K9P  416   // padded
#define PW   66    // haloed width
#define PP   4356  // 66*66

__device__ __forceinline__ float bn_gelu(float x, float g, float b, float m, float v) {
  float y = (x - m) * (g * rsqrtf(v + 1e-5f)) + b;
  return 0.5f * y * (1.0f + erff(y * 0.70710678118654752f));
}
__device__ __forceinline__ float sigmoidf(float x) { return 1.0f / (1.0f + expf(-x)); }

// ---------------------------------------------------------------------------
// Prep: zero-pad a weight matrix (M x K) into (48 x Kpad)
// ---------------------------------------------------------------------------
__global__ void k_padw(const float* __restrict__ w, float* __restrict__ wp, int M, int K,
                       int Kpad) {
  int idx = blockIdx.x * 256 + threadIdx.x;
  if (idx >= 48 * Kpad) return;
  int m = idx / Kpad, k = idx - m * Kpad;
  wp[idx] = (m < M && k < K) ? w[m * K + k] : 0.f;
}

__global__ void k_padvec(const float* __restrict__ v, float* __restrict__ vp, int n) {
  int i = threadIdx.x;
  if (i < 48) vp[i] = (i < n) ? v[i] : 0.f;
}

// ---------------------------------------------------------------------------
// Prep: build zero-halo copy of x_in: xp[b][c][66][66]
// ---------------------------------------------------------------------------
__global__ void k_halo(const float* __restrict__ xin, float* __restrict__ xp) {
  int idx = blockIdx.x * 256 + threadIdx.x;
  if (idx >= BB * CC * PP) return;
  int col = idx % PW;
  int row = (idx / PW) % PW;
  int bc = idx / PP;
  int h = row - 1, w = col - 1;
  float v = 0.f;
  if (h >= 0 && h < HH && w >= 0 && w < WWi) v = xin[(size_t)bc * HWp + h * WWi + w];
  xp[idx] = v;
}

// ---------------------------------------------------------------------------
// WMMA GEMM:  xin[b,m,n] = sum_k W[m,k] * X[b,k,n]     (proj_in)
// Wp is zero-padded 48x48.
// ---------------------------------------------------------------------------
__global__ void k_gemm_proj_in(const float* __restrict__ X, const float* __restrict__ Wp,
                               float* __restrict__ Out) {
  int lane = threadIdx.x & 31, wave = threadIdx.x >> 5;
  int b = blockIdx.y;
  int ncol = (blockIdx.x * 4 + wave) * 16 + (lane & 15);
  int khalf = (lane >> 4) << 1;
  int mrow = lane & 15;
  const float* Xb = X + (size_t)b * CC * HWp;
  const float* A0 = Wp + mrow * 48;
  const float* A1 = Wp + (16 + mrow) * 48;
  const float* A2 = Wp + (32 + mrow) * 48;
  v8f acc0 = {}, acc1 = {}, acc2 = {};
  for (int k0 = 0; k0 < 44; k0 += 4) {
    int ka = k0 + khalf;
    v2f bv = {Xb[(size_t)ka * HWp + ncol], Xb[(size_t)(ka + 1) * HWp + ncol]};
    v2f a0 = {A0[ka], A0[ka + 1]};
    v2f a1 = {A1[ka], A1[ka + 1]};
    v2f a2 = {A2[ka], A2[ka + 1]};
    acc0 = WMMA4(a0, bv, acc0);
    acc1 = WMMA4(a1, bv, acc1);
    acc2 = WMMA4(a2, bv, acc2);
  }
  {  // tail: k = 44..47 (valid 44,45)
    int ka = 44 + khalf;
    v2f bv;
    bv.x = (ka < CC) ? Xb[(size_t)ka * HWp + ncol] : 0.f;
    bv.y = (ka + 1 < CC) ? Xb[(size_t)(ka + 1) * HWp + ncol] : 0.f;
    v2f a0 = {A0[ka], A0[ka + 1]};
    v2f a1 = {A1[ka], A1[ka + 1]};
    v2f a2 = {A2[ka], A2[ka + 1]};
    acc0 = WMMA4(a0, bv, acc0);
    acc1 = WMMA4(a1, bv, acc1);
    acc2 = WMMA4(a2, bv, acc2);
  }
  int madd = (lane >> 4) * 8;
  float* Ob = Out + (size_t)b * CC * HWp;
#pragma unroll
  for (int i = 0; i < 8; ++i) {
    int m = i + madd;
    Ob[(size_t)m * HWp + ncol] = acc0[i];
    Ob[(size_t)(16 + m) * HWp + ncol] = acc1[i];
    if (32 + m < CC) Ob[(size_t)(32 + m) * HWp + ncol] = acc2[i];
  }
}

// ---------------------------------------------------------------------------
// WMMA implicit-GEMM dense 3x3 conv (loc3): K = C*9, haloed input, BN+GELU
// ---------------------------------------------------------------------------
__device__ __forceinline__ float tapP(const float* __restrict__ base, int k) {
  if (k >= K9) return 0.f;
  int c = k / 9, t = k - c * 9;
  return base[(size_t)c * PP + (t / 3) * PW + (t - (t / 3) * 3)];
}

__global__ void k_loc3(const float* __restrict__ Xp, const float* __restrict__ Wp,
                       const float* __restrict__ gg, const float* __restrict__ bb,
                       const float* __restrict__ mm, const float* __restrict__ vv,
                       float* __restrict__ branches) {
  int lane = threadIdx.x & 31, wave = threadIdx.x >> 5;
  int b = blockIdx.y;
  int ncol = (blockIdx.x * 4 + wave) * 16 + (lane & 15);
  int h = ncol >> 6, w = ncol & 63;
  int khalf = (lane >> 4) << 1;
  int mrow = lane & 15;
  const float* base = Xp + (size_t)b * CC * PP + h * PW + w;  // tap: + c*PP + dy*PW + dx
  const float* A0 = Wp + mrow * K9P;
  const float* A1 = Wp + (16 + mrow) * K9P;
  const float* A2 = Wp + (32 + mrow) * K9P;
  v8f acc0 = {}, acc1 = {}, acc2 = {};
  for (int k0 = 0; k0 < 412; k0 += 4) {
    int ka = k0 + khalf;
    int c0 = ka / 9, t0 = ka - c0 * 9;
    int ka1 = ka + 1;
    int c1 = ka1 / 9, t1 = ka1 - c1 * 9;
    v2f bv = {base[(size_t)c0 * PP + (t0 / 3) * PW + (t0 - (t0 / 3) * 3)],
              base[(size_t)c1 * PP + (t1 / 3) * PW + (t1 - (t1 / 3) * 3)]};
    v2f a0 = {A0[ka], A0[ka + 1]};
    v2f a1 = {A1[ka], A1[ka + 1]};
    v2f a2 = {A2[ka], A2[ka + 1]};
    acc0 = WMMA4(a0, bv, acc0);
    acc1 = WMMA4(a1, bv, acc1);
    acc2 = WMMA4(a2, bv, acc2);
  }
  {  // tail: k = 412..415 (valid 412,413)
    int ka = 412 + khalf;
    v2f bv;
    bv.x = tapP(base, ka);
    bv.y = tapP(base, ka + 1);
    v2f a0 = {A0[ka], A0[ka + 1]};
    v2f a1 = {A1[ka], A1[ka + 1]};
    v2f a2 = {A2[ka], A2[ka + 1]};
    acc0 = WMMA4(a0, bv, acc0);
    acc1 = WMMA4(a1, bv, acc1);
    acc2 = WMMA4(a2, bv, acc2);
  }
  int madd = (lane >> 4) * 8;
  float* Ob = branches + ((size_t)b * C6 + 4 * CC) * HWp;
#pragma unroll
  for (int i = 0; i < 8; ++i) {
    int m = i + madd;
    Ob[(size_t)m * HWp + ncol] = bn_gelu(acc0[i], gg[m], bb[m], mm[m], vv[m]);
    int m1 = 16 + m;
    Ob[(size_t)m1 * HWp + ncol] = bn_gelu(acc1[i], gg[m1], bb[m1], mm[m1], vv[m1]);
    int m2 = 32 + m;
    if (m2 < CC) Ob[(size_t)m2 * HWp + ncol] = bn_gelu(acc2[i], gg[m2], bb[m2], mm[m2], vv[m2]);
  }
}

// ---------------------------------------------------------------------------
// Deformable depthwise strip conv (lh/lw/sh/sw): offsets -> bilinear gather
// ---------------------------------------------------------------------------
__global__ void k_deform(const float* __restrict__ xin,
                         const float* __restrict__ ow, const float* __restrict__ ob,
                         const float* __restrict__ wt,
                         const float* __restrict__ gg, const float* __restrict__ bb,
                         const float* __restrict__ mm, const float* __restrict__ vv,
                         float* __restrict__ branches, int branchOff, int isVert) {
  __shared__ float s_off[CC * WWi];
  __shared__ float s_wx[KT * WWi];
  __shared__ float s_wy[KT * WWi];
  __shared__ int   s_x0[KT * WWi];
  __shared__ int   s_y0[KT * WWi];
  int h = blockIdx.x, b = blockIdx.y;
  int tid = threadIdx.x;
  const float* Xb = xin + (size_t)b * CC * HWp;

  // Stage 1: depthwise 3x3 offset conv (2K == C channels), row h
  for (int t = tid; t < CC * WWi; t += 256) {
    int j = t >> 6, w = t & 63;
    const float* Xc = Xb + (size_t)j * HWp;
    float acc = ob[j];
#pragma unroll
    for (int dy = -1; dy <= 1; ++dy) {
      int hy = h + dy;
      if (hy < 0 || hy >= HH) continue;
#pragma unroll
      for (int dx = -1; dx <= 1; ++dx) {
        int wx = w + dx;
        if (wx < 0 || wx >= WWi) continue;
        acc += Xc[hy * WWi + wx] * ow[j * 9 + (dy + 1) * 3 + (dx + 1)];
      }
    }
    s_off[t] = acc;
  }
  __syncthreads();

  // Stage 2: per-(tap, col) bilinear coordinates (shared across channels)
  const float sc = 64.0f / 63.0f;
  for (int t = tid; t < KT * WWi; t += 256) {
    int k = t >> 6, w = t & 63;
    float offx = s_off[k * WWi + w];
    float offy = s_off[(KT + k) * WWi + w];
    float px = (float)w + (isVert ? 0.0f : (float)(k - 11)) + offx;
    float py = (float)h + (isVert ? (float)(k - 11) : 0.0f) + offy;
    float ix = px * sc - 0.5f;
    float iy = py * sc - 0.5f;
    float x0 = floorf(ix), y0 = floorf(iy);
    s_wx[t] = ix - x0;
    s_wy[t] = iy - y0;
    s_x0[t] = (int)x0;
    s_y0[t] = (int)y0;
  }
  __syncthreads();

  // Stage 3: 4-corner gather (clip-then-mask), tap-weighted sum, BN + GELU
  for (int t = tid; t < CC * WWi; t += 256) {
    int c = t >> 6, w = t & 63;
    const float* Xc = Xb + (size_t)c * HWp;
    float acc = 0.f;
    for (int k = 0; k < KT; ++k) {
      int q = k * WWi + w;
      int x0 = s_x0[q], y0 = s_y0[q];
      float wx = s_wx[q], wy = s_wy[q];
      float s = 0.f;
#pragma unroll
      for (int cy = 0; cy < 2; ++cy) {
        int yi = y0 + cy;
        float wyv = cy ? wy : (1.f - wy);
        bool vy = (yi >= 0) && (yi < HH);
        int yc = min(max(yi, 0), HH - 1);
#pragma unroll
        for (int cx = 0; cx < 2; ++cx) {
          int xi = x0 + cx;
          float wxv = cx ? wx : (1.f - wx);
          bool vx = (xi >= 0) && (xi < WWi);
          int xc = min(max(xi, 0), WWi - 1);
          float val = Xc[yc * WWi + xc];
          s += val * ((vy && vx) ? (wyv * wxv) : 0.f);
        }
      }
      acc += s * wt[c * KT + k];
    }
    branches[((size_t)b * C6 + branchOff + c) * HWp + h * WWi + w] =
        bn_gelu(acc, gg[c], bb[c], mm[c], vv[c]);
  }
}

// ---------------------------------------------------------------------------
// loc5: depthwise 3x3, dilation 2, pad 2, BN+GELU
// ---------------------------------------------------------------------------
__global__ void k_loc5(const float* __restrict__ xin, const float* __restrict__ w5,
                       const float* __restrict__ gg, const float* __restrict__ bb,
                       const float* __restrict__ mm, const float* __restrict__ vv,
                       float* __restrict__ branches) {
  int idx = blockIdx.x * 256 + threadIdx.x;
  if (idx >= BB * CC * HWp) return;
  int w = idx & 63, h = (idx >> 6) & 63;
  int c = (idx >> 12) % CC;
  int b = idx / (CC * HWp);
  const float* Xc = xin + ((size_t)b * CC + c) * HWp;
  float acc = 0.f;
#pragma unroll
  for (int dy = 0; dy < 3; ++dy) {
    int hy = h + 2 * (dy - 1);
    if (hy < 0 || hy >= HH) continue;
#pragma unroll
    for (int dx = 0; dx < 3; ++dx) {
      int wx = w + 2 * (dx - 1);
      if (wx < 0 || wx >= WWi) continue;
      acc += Xc[hy * WWi + wx] * w5[c * 9 + dy * 3 + dx];
    }
  }
  branches[((size_t)b * C6 + 5 * CC + c) * HWp + h * WWi + w] =
      bn_gelu(acc, gg[c], bb[c], mm[c], vv[c]);
}

// ---------------------------------------------------------------------------
// Pupil gate: 1ch -> 11ch relu conv, then 11ch -> 1ch sigmoid conv
// ---------------------------------------------------------------------------
__global__ void k_pg1(const float* __restrict__ heat, const float* __restrict__ w1,
                      const float* __restrict__ b1, float* __restrict__ pgh) {
  int idx = blockIdx.x * 256 + threadIdx.x;
  if (idx >= BB * 11 * HWp) return;
  int w = idx & 63, h = (idx >> 6) & 63;
  int oc = (idx >> 12) % 11;
  int b = idx / (11 * HWp);
  const float* Hb = heat + (size_t)b * HWp;
  float acc = b1[oc];
#pragma unroll
  for (int dy = -1; dy <= 1; ++dy) {
    int hy = h + dy;
    if (hy < 0 || hy >= HH) continue;
#pragma unroll
    for (int dx = -1; dx <= 1; ++dx) {
      int wx = w + dx;
      if (wx < 0 || wx >= WWi) continue;
      acc += Hb[hy * WWi + wx] * w1[oc * 9 + (dy + 1) * 3 + (dx + 1)];
    }
  }
  pgh[idx] = fmaxf(acc, 0.f);
}

__global__ void k_pg2(const float* __restrict__ pgh, const float* __restrict__ w2,
                      const float* __restrict__ b2, float* __restrict__ spatial) {
  int idx = blockIdx.x * 256 + threadIdx.x;
  if (idx >= BB * HWp) return;
  int w = idx & 63, h = (idx >> 6) & 63;
  int b = idx / HWp;
  float acc = b2[0];
  for (int c = 0; c < 11; ++c) {
    const float* Hc = pgh + ((size_t)b * 11 + c) * HWp;
#pragma unroll
    for (int dy = -1; dy <= 1; ++dy) {
      int hy = h + dy;
      if (hy < 0 || hy >= HH) continue;
#pragma unroll
      for (int dx = -1; dx <= 1; ++dx) {
        int wx = w + dx;
        if (wx < 0 || wx >= WWi) continue;
        acc += Hc[hy * WWi + wx] * w2[c * 9 + (dy + 1) * 3 + (dx + 1)];
      }
    }
  }
  spatial[idx] = sigmoidf(acc);
}

// ---------------------------------------------------------------------------
// Global average pool per (b,c)
// ---------------------------------------------------------------------------
__global__ void k_gv(const float* __restrict__ xin, float* __restrict__ gv) {
  __shared__ float red[256];
  int bc = blockIdx.x;
  const float* Xc = xin + (size_t)bc * HWp;
  float s = 0.f;
  for (int i = threadIdx.x; i < HWp; i += 256) s += Xc[i];
  red[threadIdx.x] = s;
  __syncthreads();
  for (int st = 128; st > 0; st >>= 1) {
    if (threadIdx.x < st) red[threadIdx.x] += red[threadIdx.x + st];
    __syncthreads();
  }
  if (threadIdx.x == 0) gv[bc] = red[0] * (1.0f / (float)HWp);
}

// ---------------------------------------------------------------------------
// Modal conditioning MLP -> sigmoid(scale), shift  (tiny; one block)
// ---------------------------------------------------------------------------
__global__ void k_mlp(const float* __restrict__ gv,
                      const float* __restrict__ w1, const float* __restrict__ b1,
                      const float* __restrict__ w2, const float* __restrict__ b2,
                      const float* __restrict__ msw, const float* __restrict__ msb,
                      const float* __restrict__ mshw, const float* __restrict__ mshb,
                      float* __restrict__ sscale, float* __restrict__ sshift) {
  const int C2 = 23;
  __shared__ float s_h[BB * C2];
  __shared__ float s_mc[BB * CC];
  int tid = threadIdx.x;
  for (int t = tid; t < BB * C2; t += 256) {
    int b = t / C2, j = t % C2;
    float s = b1[j];
    for (int c = 0; c < CC; ++c) s += gv[b * CC + c] * w1[j * CC + c];
    s_h[t] = fmaxf(s, 0.f);
  }
  __syncthreads();
  for (int t = tid; t < BB * CC; t += 256) {
    int b = t / CC, i = t % CC;
    float s = b2[i];
    for (int j = 0; j < C2; ++j) s += s_h[b * C2 + j] * w2[i * C2 + j];
    s_mc[t] = s;
  }
  __syncthreads();
  for (int t = tid; t < BB * CC; t += 256) {
    int b = t / CC, c = t % CC;
    float sc2 = msb[c], sh = mshb[c];
    for (int i = 0; i < CC; ++i) {
      float m = s_mc[b * CC + i];
      sc2 += m * msw[c * CC + i];
      sh += m * mshw[c * CC + i];
    }
    sscale[t] = sigmoidf(sc2);
    sshift[t] = sh;
  }
}

// ---------------------------------------------------------------------------
// Gate GEMM (K=276): acc = gate_b + gate_w * (branches * (1+spatial));
// epilogue: outbuf = x_in * (sigmoid(acc)*sscale + sshift)
// Wp padded 48x276, gbp padded 48.
// ---------------------------------------------------------------------------
__global__ void k_gate(const float* __restrict__ branches, const float* __restrict__ spatial,
                       const float* __restrict__ xin, const float* __restrict__ Wp,
                       const float* __restrict__ gbp, const float* __restrict__ sscale,
                       const float* __restrict__ sshift, float* __restrict__ outbuf) {
  int lane = threadIdx.x & 31, wave = threadIdx.x >> 5;
  int b = blockIdx.y;
  int ncol = (blockIdx.x * 4 + wave) * 16 + (lane & 15);
  int khalf = (lane >> 4) << 1;
  int mrow = lane & 15;
  int madd = (lane >> 4) * 8;
  const float* Br = branches + (size_t)b * C6 * HWp;
  const float* A0 = Wp + mrow * C6;
  const float* A1 = Wp + (16 + mrow) * C6;
  const float* A2 = Wp + (32 + mrow) * C6;
  float sp = 1.0f + spatial[(size_t)b * HWp + ncol];
  v8f acc0, acc1, acc2;
#pragma unroll
  for (int i = 0; i < 8; ++i) {
    int m = i + madd;
    acc0[i] = gbp[m];
    acc1[i] = gbp[16 + m];
    acc2[i] = gbp[32 + m];
  }
  for (int k0 = 0; k0 < C6; k0 += 4) {  // 276 % 4 == 0, no tail
    int ka = k0 + khalf;
    if (k0 + 8 < C6) __builtin_prefetch(&Br[(size_t)(k0 + 8) * HWp + ncol], 0, 1);
    v2f bv = {Br[(size_t)ka * HWp + ncol] * sp, Br[(size_t)(ka + 1) * HWp + ncol] * sp};
    v2f a0 = {A0[ka], A0[ka + 1]};
    v2f a1 = {A1[ka], A1[ka + 1]};
    v2f a2 = {A2[ka], A2[ka + 1]};
    acc0 = WMMA4(a0, bv, acc0);
    acc1 = WMMA4(a1, bv, acc1);
    acc2 = WMMA4(a2, bv, acc2);
  }
  const float* Xb = xin + (size_t)b * CC * HWp;
  float* Ob = outbuf + (size_t)b * CC * HWp;
#pragma unroll
  for (int i = 0; i < 8; ++i) {
    int ms[3] = {i + madd, 16 + i + madd, 32 + i + madd};
    float av[3] = {acc0[i], acc1[i], acc2[i]};
#pragma unroll
    for (int t = 0; t < 3; ++t) {
      int m = ms[t];
      if (m < CC) {
        float g = sigmoidf(av[t]);
        float mod = g * sscale[b * CC + m] + sshift[b * CC + m];
        Ob[(size_t)m * HWp + ncol] = Xb[(size_t)m * HWp + ncol] * mod;
      }
    }
  }
}

// ---------------------------------------------------------------------------
// proj_out GEMM with residual: D = shortcut + proj_out_w * outbuf
// ---------------------------------------------------------------------------
__global__ void k_gemm_proj_out(const float* __restrict__ X, const float* __restrict__ Wp,
                                const float* __restrict__ shortcut, float* __restrict__ Out) {
  int lane = threadIdx.x & 31, wave = threadIdx.x >> 5;
  int b = blockIdx.y;
  int ncol = (blockIdx.x * 4 + wave) * 16 + (lane & 15);
  int khalf = (lane >> 4) << 1;
  int mrow = lane & 15;
  int madd = (lane >> 4) * 8;
  const float* Xb = X + (size_t)b * CC * HWp;
  const float* Sb = shortcut + (size_t)b * CC * HWp;
  const float* A0 = Wp + mrow * 48;
  const float* A1 = Wp + (16 + mrow) * 48;
  const float* A2 = Wp + (32 + mrow) * 48;
  v8f acc0, acc1, acc2;
#pragma unroll
  for (int i = 0; i < 8; ++i) {
    int m = i + madd;
    acc0[i] = Sb[(size_t)m * HWp + ncol];
    acc1[i] = Sb[(size_t)(16 + m) * HWp + ncol];
    acc2[i] = (32 + m < CC) ? Sb[(size_t)(32 + m) * HWp + ncol] : 0.f;
  }
  for (int k0 = 0; k0 < 44; k0 += 4) {
    int ka = k0 + khalf;
    v2f bv = {Xb[(size_t)ka * HWp + ncol], Xb[(size_t)(ka + 1) * HWp + ncol]};
    v2f a0 = {A0[ka], A0[ka + 1]};
    v2f a1 = {A1[ka], A1[ka + 1]};
    v2f a2 = {A2[ka], A2[ka + 1]};
    acc0 = WMMA4(a0, bv, acc0);
    acc1 = WMMA4(a1, bv, acc1);
    acc2 = WMMA4(a2, bv, acc2);
  }
  {  // tail: k = 44..47 (valid 44,45)
    int ka = 44 + khalf;
    v2f bv;
    bv.x = (ka < CC) ? Xb[(size_t)ka * HWp + ncol] : 0.f;
    bv.y = (ka + 1 < CC) ? Xb[(size_t)(ka + 1) * HWp + ncol] : 0.f;
    v2f a0 = {A0[ka], A0[ka + 1]};
    v2f a1 = {A1[ka], A1[ka + 1]};
    v2f a2 = {A2[ka], A2[ka + 1]};
    acc0 = WMMA4(a0, bv, acc0);
    acc1 = WMMA4(a1, bv, acc1);
    acc2 = WMMA4(a2, bv, acc2);
  }
  float* Ob = Out + (size_t)b * CC * HWp;
#pragma unroll
  for (int i = 0; i < 8; ++i) {
    int m = i + madd;
    Ob[(size_t)m * HWp + ncol] = acc0[i];
    Ob[(size_t)(16 + m) * HWp + ncol] = acc1[i];
    if (32 + m < CC) Ob[(size_t)(32 + m) * HWp + ncol] = acc2[i];
  }
}

// ---------------------------------------------------------------------------
extern "C" void kernel_launch(void* const* d_in, const int* in_sizes, int n_in,
                              void* d_out, int out_size, void* d_ws, size_t ws_size,
                              hipStream_t stream) {
  (void)in_sizes; (void)n_in; (void)out_size; (void)ws_size;
  const float* x          = (const float*)d_in[0];
  const float* heat       = (const float*)d_in[1];
  const float* proj_in_w  = (const float*)d_in[2];
  // lh=3, lw=10, sh=17, sw=24 ; within: +0 ow,+1 ob,+2 w,+3 g,+4 b,+5 m,+6 v
  const float* loc3_w = (const float*)d_in[31];
  const float* loc3_g = (const float*)d_in[32];
  const float* loc3_b = (const float*)d_in[33];
  const float* loc3_m = (const float*)d_in[34];
  const float* loc3_v = (const float*)d_in[35];
  const float* loc5_w = (const float*)d_in[36];
  const float* loc5_g = (const float*)d_in[37];
  const float* loc5_b = (const float*)d_in[38];
  const float* loc5_m = (const float*)d_in[39];
  const float* loc5_v = (const float*)d_in[40];
  const float* pg_w1  = (const float*)d_in[41];
  const float* pg_b1  = (const float*)d_in[42];
  const float* pg_w2  = (const float*)d_in[43];
  const float* pg_b2  = (const float*)d_in[44];
  const float* gate_w = (const float*)d_in[45];
  const float* gate_b = (const float*)d_in[46];
  const float* me_w1  = (const float*)d_in[47];
  const float* me_b1  = (const float*)d_in[48];
  const float* me_w2  = (const float*)d_in[49];
  const float* me_b2  = (const float*)d_in[50];
  const float* ms_w   = (const float*)d_in[51];
  const float* ms_b   = (const float*)d_in[52];
  const float* msh_w  = (const float*)d_in[53];
  const float* msh_b  = (const float*)d_in[54];
  const float* proj_out_w = (const float*)d_in[55];

  const size_t BCHW = (size_t)BB * CC * HWp;
  float* ws       = (float*)d_ws;
  float* x_in     = ws;
  float* branches = x_in + BCHW;                   // 6*BCHW
  float* spatial  = branches + 6 * BCHW;           // BB*HWp
  float* pgh      = spatial + (size_t)BB * HWp;    // BB*11*HWp
  float* outbuf   = pgh + (size_t)BB * 11 * HWp;   // BCHW
  float* gvbuf    = outbuf + BCHW;                 // BB*CC
  float* sscale   = gvbuf + BB * CC;               // BB*CC
  float* sshift   = sscale + BB * CC;              // BB*CC
  float* xpad     = sshift + BB * CC;              // BB*CC*PP
  float* wp_pin   = xpad + (size_t)BB * CC * PP;   // 48*48
  float* wp_loc3  = wp_pin + 48 * 48;              // 48*416
  float* wp_gate  = wp_loc3 + 48 * K9P;            // 48*276
  float* wp_pout  = wp_gate + 48 * C6;             // 48*48
  float* gbp      = wp_pout + 48 * 48;             // 48

  dim3 gemm_grid(HWp / 64, BB);

  // --- prep: zero-padded weights ---
  k_padw<<<(48 * 48 + 255) / 256, 256, 0, stream>>>(proj_in_w, wp_pin, CC, CC, 48);
  k_padw<<<(48 * K9P + 255) / 256, 256, 0, stream>>>(loc3_w, wp_loc3, CC, K9, K9P);
  k_padw<<<(48 * C6 + 255) / 256, 256, 0, stream>>>(gate_w, wp_gate, CC, C6, C6);
  k_padw<<<(48 * 48 + 255) / 256, 256, 0, stream>>>(proj_out_w, wp_pout, CC, CC, 48);
  k_padvec<<<1, 64, 0, stream>>>(gate_b, gbp, CC);

  // --- main pipeline ---
  k_gemm_proj_in<<<gemm_grid, 128, 0, stream>>>(x, wp_pin, x_in);
  k_halo<<<((int)(BB * CC * PP) + 255) / 256, 256, 0, stream>>>(x_in, xpad);

  const int base[4] = {3, 10, 17, 24};
  const int isV[4] = {1, 0, 1, 0};
  for (int br = 0; br < 4; ++br) {
    int i0 = base[br];
    k_deform<<<dim3(HH, BB), 256, 0, stream>>>(
        x_in, (const float*)d_in[i0], (const float*)d_in[i0 + 1], (const float*)d_in[i0 + 2],
        (const float*)d_in[i0 + 3], (const float*)d_in[i0 + 4], (const float*)d_in[i0 + 5],
        (const float*)d_in[i0 + 6], branches, br * CC, isV[br]);
  }
  k_loc3<<<gemm_grid, 128, 0, stream>>>(xpad, wp_loc3, loc3_g, loc3_b, loc3_m, loc3_v, branches);
  k_loc5<<<(BB * CC * HWp + 255) / 256, 256, 0, stream>>>(x_in, loc5_w, loc5_g, loc5_b, loc5_m,
                                                          loc5_v, branches);
  k_pg1<<<(BB * 11 * HWp + 255) / 256, 256, 0, stream>>>(heat, pg_w1, pg_b1, pgh);
  k_pg2<<<(BB * HWp + 255) / 256, 256, 0, stream>>>(pgh, pg_w2, pg_b2, spatial);
  k_gv<<<BB * CC, 256, 0, stream>>>(x_in, gvbuf);
  k_mlp<<<1, 256, 0, stream>>>(gvbuf, me_w1, me_b1, me_w2, me_b2, ms_w, ms_b, msh_w, msh_b,
                               sscale, sshift);
  k_gate<<<gemm_grid, 128, 0, stream>>>(branches, spatial, x_in, wp_gate, gbp, sscale, sshift,
                                        outbuf);
  k_gemm_proj_out<<<gemm_grid, 128, 0, stream>>>(outbuf, wp_pout, x, (float*)d_out);
}